// Decoder_49039936586036
// MI455X (gfx1250) — compile-verified
//
#include <hip/hip_runtime.h>
#include <stdint.h>

// ---------------------------------------------------------------------------
// 27-state Viterbi decode, B=256, L=4096.
//   fwd: 1 wave per batch, lane = state. Per step:
//        - 5-hop butterfly (max, min-index) reduction  -> (-100) fallback
//        - <=5 shuffle gathers from fixed sparse predecessor lists
//        - exact lexicographic combine == jnp.argmax first-occurrence ties
//        - history byte per (state, step) staged in LDS, async-flushed (B128)
//        - emissions double-buffered into LDS via async global->LDS DMA
//   bwd: 1 wave per batch, history tiles async-prefetched to LDS, serial
//        pointer-chase through LDS, coalesced 32-wide int32 output stores.
// Workspace layout in d_ws: hist u8[B][L][32]  (32 MB), then endtag int[B].
// ---------------------------------------------------------------------------

#define LLEN   4096
#define BATCH  256
#define NTAGS  27
#define TILE   128
#define NTILES (LLEN / TILE)
#define HPAD   32
#define NEGF   (-3.0e38f)

#if __has_builtin(__builtin_amdgcn_global_load_async_to_lds_b128) && \
    __has_builtin(__builtin_amdgcn_global_store_async_from_lds_b128)
#define ASYNC_OK 1
#else
#define ASYNC_OK 0
#endif

#if ASYNC_OK
typedef int v4i __attribute__((vector_size(16)));
typedef v4i __attribute__((address_space(1))) g_v4i;   // global
typedef v4i __attribute__((address_space(3))) l_v4i;   // LDS

__device__ __forceinline__ void async_g2l_b128(const void* g, void* l) {
  __builtin_amdgcn_global_load_async_to_lds_b128(
      (g_v4i*)(uintptr_t)g,
      (l_v4i*)(uint32_t)(uintptr_t)l,
      0, 0);
}
__device__ __forceinline__ void async_l2g_b128(void* g, const void* l) {
  __builtin_amdgcn_global_store_async_from_lds_b128(
      (g_v4i*)(uintptr_t)g,
      (l_v4i*)(uint32_t)(uintptr_t)l,
      0, 0);
}
#define WAIT_ASYNC() asm volatile("s_wait_asynccnt 0x0" ::: "memory")
#define WAIT_DS()    asm volatile("s_wait_dscnt 0x0" ::: "memory")
#endif

// Sparse predecessor lists (ascending index, pad = 31 -> lane31 holds -inf).
__constant__ int PRED0[32] = {25, 0, 1, 2, 3,  26, 5, 6, 7, 8,
                              25,10,11,12,13,  26,15,16,17,18,
                              31,20,21,22,23,   9, 4,
                              31,31,31,31,31};
__constant__ int PRED1[32] = {31,31,31,31, 4,  31,31,31,31, 9,
                              31,31,31,31,14,  31,31,31,31,19,
                              31,31,31,31,24,  19,14,
                              31,31,31,31,31};
__constant__ int PRED2[32] = {31,31,31,31,31,  31,31,31,31,31,
                              31,31,31,31,31,  31,31,31,31,31,
                              31,31,31,31,31,  24,24,
                              31,31,31,31,31};
__constant__ int PRED3[32] = {31,31,31,31,31,  31,31,31,31,31,
                              31,31,31,31,31,  31,31,31,31,31,
                              31,31,31,31,31,  25,25,
                              31,31,31,31,31};
__constant__ int PRED4[32] = {31,31,31,31,31,  31,31,31,31,31,
                              31,31,31,31,31,  31,31,31,31,31,
                              31,31,31,31,31,  26,26,
                              31,31,31,31,31};
// emission channel per state (repeats 10,10,5,1,1)
__constant__ int CHAN_T[32] = {0,0,0,0,0, 0,0,0,0,0, 1,1,1,1,1, 1,1,1,1,1,
                               2,2,2,2,2, 3,4, 0,0,0,0,0};
__constant__ float START_T[32] = {0.f,-100.f,-100.f,-100.f,-100.f,
                                  0.f,-100.f,-100.f,-100.f,-100.f,
                                  0.f,-100.f,-100.f,-100.f,-100.f,
                                  0.f,-100.f,-100.f,-100.f,-100.f,
                                  0.f,-100.f,-100.f,-100.f,-100.f,
                                  0.f, 0.f, 0.f,0.f,0.f,0.f,0.f};
__constant__ float END_T[32]   = {-100.f,-100.f,-100.f,-100.f,0.f,
                                  -100.f,-100.f,-100.f,-100.f,0.f,
                                  -100.f,-100.f,-100.f,-100.f,0.f,
                                  -100.f,-100.f,-100.f,-100.f,0.f,
                                  -100.f,-100.f,-100.f,-100.f,0.f,
                                  0.f, 0.f, 0.f,0.f,0.f,0.f,0.f};

__global__ __launch_bounds__(32)
void viterbi_fwd(const float* __restrict__ em,
                 const unsigned char* __restrict__ mask,
                 unsigned char* __restrict__ hist,
                 int* __restrict__ endtag)
{
  const int  b      = blockIdx.x;
  const int  lane   = threadIdx.x;
  const bool active = lane < NTAGS;

  const int   chan = CHAN_T[lane];
  const int   p0 = PRED0[lane], p1 = PRED1[lane], p2 = PRED2[lane],
              p3 = PRED3[lane], p4 = PRED4[lane];
  const float st = START_T[lane];
  const float en = END_T[lane];

  const float*         emB   = em   + (size_t)b * 5 * LLEN;
  const unsigned char* mkB   = mask + (size_t)b * LLEN;
  unsigned char*       histB = hist + (size_t)b * LLEN * HPAD;

#if ASYNC_OK
  __shared__ float         sEm[2][5 * TILE];
  __shared__ unsigned char sHist[2][TILE * HPAD];

  auto load_em_tile = [&](int k) {
    const int buf = k & 1;
    #pragma unroll
    for (int c = 0; c < 5; ++c)
      async_g2l_b128(emB + (size_t)c * LLEN + k * TILE + lane * 4,
                     &sEm[buf][c * TILE + lane * 4]);
  };
  auto store_hist_tile = [&](int k) {
    const int buf = k & 1;
    unsigned char* g = histB + (size_t)k * TILE * HPAD;
    #pragma unroll
    for (int i = 0; i < (TILE * HPAD) / (32 * 16); ++i)   // 8 x B128 per lane
      async_l2g_b128(g + i * 512 + lane * 16, &sHist[buf][i * 512 + lane * 16]);
  };

  load_em_tile(0);
  WAIT_ASYNC();
  load_em_tile(1);
#endif

  float s;
  {
#if ASYNC_OK
    float e0 = sEm[0][chan * TILE + 0];
#else
    float e0 = emB[(size_t)chan * LLEN + 0];
#endif
    s = active ? (st + e0) : NEGF;
  }

  for (int t = 1; t < LLEN; ++t) {
#if ASYNC_OK
    if ((t & (TILE - 1)) == 0) {
      const int k = t >> 7;              // tile now being consumed
      WAIT_DS();                         // hist bytes of tile k-1 in LDS
      WAIT_ASYNC();                      // em tile k loaded, old flush done
      store_hist_tile(k - 1);
      if (k + 1 < NTILES) load_em_tile(k + 1);
    }
#endif
    // (max score, lowest index) over all 27 states — wave32 butterfly
    float gv = s;
    int   gi = lane;
    #pragma unroll
    for (int off = 16; off; off >>= 1) {
      float ov = __shfl_xor(gv, off, 32);
      int   oi = __shfl_xor(gi, off, 32);
      bool  tk = (ov > gv) || ((ov == gv) && (oi < gi));
      gv = tk ? ov : gv;
      gi = tk ? oi : gi;
    }
    // best allowed predecessor (ascending index, strict > keeps first-max)
    float bv = NEGF;
    int   bi = 31;
    {
      float v;
      v = __shfl(s, p0, 32); if (v > bv) { bv = v; bi = p0; }
      v = __shfl(s, p1, 32); if (v > bv) { bv = v; bi = p1; }
      v = __shfl(s, p2, 32); if (v > bv) { bv = v; bi = p2; }
      v = __shfl(s, p3, 32); if (v > bv) { bv = v; bi = p3; }
      v = __shfl(s, p4, 32); if (v > bv) { bv = v; bi = p4; }
    }
    // exact -100 fallback (lexicographic combine reproduces jnp.argmax ties)
    {
      float fv = gv - 100.0f;
      bool  tk = (fv > bv) || ((fv == bv) && (gi < bi));
      if (tk) { bv = fv; bi = gi; }
    }
#if ASYNC_OK
    float e = sEm[(t >> 7) & 1][chan * TILE + (t & (TILE - 1))];
    sHist[(t >> 7) & 1][(t & (TILE - 1)) * HPAD + lane] = (unsigned char)bi;
#else
    float e = emB[(size_t)chan * LLEN + t];
    histB[(size_t)t * HPAD + lane] = (unsigned char)bi;
#endif
    const bool mk = mkB[t] != 0;
    if (active && mk) s = bv + e;
  }

#if ASYNC_OK
  WAIT_DS();
  WAIT_ASYNC();
  store_hist_tile(NTILES - 1);
  WAIT_ASYNC();
#endif

  // end tag: argmax(score + end), first-occurrence ties
  float ev = active ? (s + en) : NEGF;
  int   ei = lane;
  #pragma unroll
  for (int off = 16; off; off >>= 1) {
    float ov = __shfl_xor(ev, off, 32);
    int   oi = __shfl_xor(ei, off, 32);
    bool  tk = (ov > ev) || ((ov == ev) && (oi < ei));
    ev = tk ? ov : ev;
    ei = tk ? oi : ei;
  }
  if (lane == 0) endtag[b] = ei;
}

__global__ __launch_bounds__(32)
void viterbi_bwd(const unsigned char* __restrict__ hist,
                 const int* __restrict__ endtag,
                 const unsigned char* __restrict__ mask,
                 int* __restrict__ out)
{
  const int b    = blockIdx.x;
  const int lane = threadIdx.x;
  const unsigned char* histB = hist + (size_t)b * LLEN * HPAD;
  const unsigned char* mkB   = mask + (size_t)b * LLEN;

  int acc = 0;
  for (int t = lane; t < LLEN; t += 32) acc += mkB[t];
  #pragma unroll
  for (int off = 16; off; off >>= 1) acc += __shfl_xor(acc, off, 32);
  const int seq_end = acc - 1;
  const int etag    = endtag[b];

#if ASYNC_OK
  __shared__ unsigned char sT[2][TILE * HPAD];
  auto load_hist_tile = [&](int k) {
    const int buf = k & 1;
    const unsigned char* g = histB + (size_t)k * TILE * HPAD;
    #pragma unroll
    for (int i = 0; i < (TILE * HPAD) / (32 * 16); ++i)
      async_g2l_b128(g + i * 512 + lane * 16, &sT[buf][i * 512 + lane * 16]);
  };
  load_hist_tile(NTILES - 1);
#endif

  int bt  = 0;
  int myv = 0;
  // tile k serves reads of history rows r = t+1 in [k*TILE, (k+1)*TILE-1]
  for (int k = NTILES - 1; k >= 0; --k) {
#if ASYNC_OK
    WAIT_ASYNC();
    if (k > 0) load_hist_tile(k - 1);
#endif
    const int thi = (k == NTILES - 1) ? (LLEN - 1) : ((k + 1) * TILE - 2);
    const int tlo = (k == 0) ? 0 : (k * TILE - 1);
    for (int t = thi; t >= tlo; --t) {
      int v;
      if (t == seq_end)        v = etag;
      else if (t == LLEN - 1)  v = 0;
      else {
        const int r = t + 1;
#if ASYNC_OK
        v = sT[k & 1][(r & (TILE - 1)) * HPAD + bt];
#else
        v = histB[(size_t)r * HPAD + bt];
#endif
      }
      bt = v;
      const int mp = (bt < 25) ? (bt / 5) : (bt - 20);   // mapping[]
      if ((t & 31) == lane) myv = mp;
      if ((t & 31) == 0) out[(size_t)b * LLEN + t + lane] = myv;  // coalesced
    }
  }
}

extern "C" void kernel_launch(void* const* d_in, const int* in_sizes, int n_in,
                              void* d_out, int out_size, void* d_ws, size_t ws_size,
                              hipStream_t stream)
{
  (void)in_sizes; (void)n_in; (void)out_size; (void)ws_size;
  const float*         em   = (const float*)d_in[0];
  const unsigned char* mask = (const unsigned char*)d_in[1];
  unsigned char* hist   = (unsigned char*)d_ws;                 // 32 MB
  int*           endtag = (int*)((unsigned char*)d_ws +
                                 (size_t)BATCH * LLEN * HPAD);  // + 1 KB
  int* out = (int*)d_out;

  viterbi_fwd<<<BATCH, 32, 0, stream>>>(em, mask, hist, endtag);
  viterbi_bwd<<<BATCH, 32, 0, stream>>>(hist, endtag, mask, out);
}